// Encoder_6262062318121
// MI455X (gfx1250) — compile-verified
//
#include <hip/hip_runtime.h>
#include <math.h>

#define SS 8
#define LL 512
#define DD 768
#define HH 12
#define EE 32
#define MM 3
#define RR 64
#define KK 51      // L * 10 / 100
#define D2 1536
#define EPSF 1e-12f

typedef __attribute__((ext_vector_type(2))) float v2f;
typedef __attribute__((ext_vector_type(8))) float v8f;

// ---------------- tiny matvecs: q = Wq@rel_cls + bq ; v = Wk^T q ; c = bk.q ---
__global__ void k_q(const float* __restrict__ Wq, const float* __restrict__ bq,
                    const float* __restrict__ rel_cls, float* __restrict__ q) {
  int d = blockIdx.x * blockDim.x + threadIdx.x;
  if (d < DD) {
    const float* w = Wq + (size_t)d * DD;
    float acc = bq[d];
    for (int j = 0; j < DD; ++j) acc += w[j] * rel_cls[j];
    q[d] = acc;
  }
}

__global__ void k_v(const float* __restrict__ Wk, const float* __restrict__ q,
                    float* __restrict__ v) {
  int j = blockIdx.x * blockDim.x + threadIdx.x;
  if (j < DD) {
    float acc = 0.0f;
    for (int d = 0; d < DD; ++d) acc += Wk[(size_t)d * DD + j] * q[d];
    v[j] = acc;
  }
}

__global__ void k_c(const float* __restrict__ bk, const float* __restrict__ q,
                    float* __restrict__ c) {
  __shared__ float red[256];
  float acc = 0.0f;
  for (int d = threadIdx.x; d < DD; d += 256) acc += bk[d] * q[d];
  red[threadIdx.x] = acc;
  __syncthreads();
  for (int s = 128; s > 0; s >>= 1) {
    if ((int)threadIdx.x < s) red[threadIdx.x] += red[threadIdx.x + s];
    __syncthreads();
  }
  if (threadIdx.x == 0) c[0] = red[0];
}

// ---------------- entity embeddings: logsumexp over M=3 mentions -------------
__global__ void k_entemb(const float* __restrict__ seq, const int* __restrict__ starts,
                         float* __restrict__ ent) {
  int idx = blockIdx.x * 256 + threadIdx.x;
  if (idx >= SS * EE * DD) return;
  int d = idx % DD;
  int se = idx / DD;
  int e = se % EE, s = se / EE;
  const int* st = starts + ((size_t)s * EE + e) * MM;
  float x0 = seq[((size_t)s * LL + st[0]) * DD + d];
  float x1 = seq[((size_t)s * LL + st[1]) * DD + d];
  float x2 = seq[((size_t)s * LL + st[2]) * DD + d];
  float mx = fmaxf(x0, fmaxf(x1, x2));
  float sm = expf(x0 - mx) + expf(x1 - mx) + expf(x2 - mx);
  ent[idx] = mx + logf(sm);
}

// ---------------- entity attention: mean over mentions, (s,e,h,l) -----------
__global__ void k_entatt(const float* __restrict__ att, const int* __restrict__ starts,
                         float* __restrict__ entA) {
  size_t idx = (size_t)blockIdx.x * 256 + threadIdx.x;
  if (idx >= (size_t)SS * EE * HH * LL) return;
  int l = (int)(idx % LL);
  size_t t = idx / LL;
  int h = (int)(t % HH); t /= HH;
  int e = (int)(t % EE);
  int s = (int)(t / EE);
  const int* st = starts + ((size_t)s * EE + e) * MM;
  const float* base = att + ((size_t)s * HH + h) * LL * LL;
  float acc = base[(size_t)st[0] * LL + l] + base[(size_t)st[1] * LL + l] +
              base[(size_t)st[2] * LL + l];
  entA[idx] = acc * (1.0f / 3.0f);
}

// ---------------- rel_att: masked softmax of seq . v * scale + c*scale ------
__global__ void k_relatt(const float* __restrict__ seq, const int* __restrict__ msk,
                         const float* __restrict__ v, const float* __restrict__ c,
                         float* __restrict__ rel) {
  int s = blockIdx.x;
  __shared__ float red[256];
  const float scale = rsqrtf((float)DD);
  const float cc = c[0];
  float logit[2];
#pragma unroll
  for (int i = 0; i < 2; ++i) {
    int l = threadIdx.x + i * 256;
    const float* row = seq + ((size_t)s * LL + l) * DD;
    float acc = cc;
    for (int j = 0; j < DD; ++j) acc += row[j] * v[j];
    acc *= scale;
    if (msk[(size_t)s * LL + l] <= 0) acc = -__builtin_inff();
    logit[i] = acc;
  }
  red[threadIdx.x] = fmaxf(logit[0], logit[1]);
  __syncthreads();
  for (int st = 128; st > 0; st >>= 1) {
    if ((int)threadIdx.x < st) red[threadIdx.x] = fmaxf(red[threadIdx.x], red[threadIdx.x + st]);
    __syncthreads();
  }
  float m = red[0];
  __syncthreads();
  float e0 = expf(logit[0] - m), e1 = expf(logit[1] - m);
  red[threadIdx.x] = e0 + e1;
  __syncthreads();
  for (int st = 128; st > 0; st >>= 1) {
    if ((int)threadIdx.x < st) red[threadIdx.x] += red[threadIdx.x + st];
    __syncthreads();
  }
  float inv = 1.0f / red[0];
  rel[(size_t)s * LL + threadIdx.x] = e0 * inv;
  rel[(size_t)s * LL + threadIdx.x + 256] = e1 * inv;
}

// ---------------- ht_att: mean-over-heads product + L1 norm ------------------
__global__ void k_htatt(const float* __restrict__ entA, const int* __restrict__ hts,
                        float* __restrict__ ht) {
  int s = blockIdx.x / RR, r = blockIdx.x % RR;
  int hi = hts[((size_t)s * RR + r) * 2 + 0];
  int ti = hts[((size_t)s * RR + r) * 2 + 1];
  __shared__ float red[256];
  float val[2];
  float asum = 0.0f;
#pragma unroll
  for (int i = 0; i < 2; ++i) {
    int l = threadIdx.x + 256 * i;
    float acc = 0.0f;
    for (int h = 0; h < HH; ++h) {
      float a = entA[(((size_t)s * EE + hi) * HH + h) * LL + l];
      float b = entA[(((size_t)s * EE + ti) * HH + h) * LL + l];
      acc += a * b;
    }
    acc *= (1.0f / (float)HH);
    val[i] = acc;
    asum += fabsf(acc);
  }
  red[threadIdx.x] = asum;
  __syncthreads();
  for (int st = 128; st > 0; st >>= 1) {
    if ((int)threadIdx.x < st) red[threadIdx.x] += red[threadIdx.x + st];
    __syncthreads();
  }
  float inv = 1.0f / fmaxf(red[0], EPSF);
#pragma unroll
  for (int i = 0; i < 2; ++i)
    ht[((size_t)s * RR + r) * LL + threadIdx.x + 256 * i] = val[i] * inv;
}

// ---------------- exact top-K threshold (bit bisection), update, renorm ------
__global__ void k_topk(const float* __restrict__ rel, float* __restrict__ ht) {
  int s = blockIdx.x / RR, r = blockIdx.x % RR;
  __shared__ unsigned int abits[LL];
  __shared__ int cnt;
  __shared__ float red[256];
  float hv[2], rv[2], av[2];
#pragma unroll
  for (int i = 0; i < 2; ++i) {
    int l = threadIdx.x + 256 * i;
    hv[i] = ht[((size_t)s * RR + r) * LL + l];
    rv[i] = rel[(size_t)s * LL + l];
    av[i] = hv[i] * rv[i];           // all non-negative -> uint-order == float-order
    abits[l] = __float_as_uint(av[i]);
  }
  __syncthreads();
  unsigned int lo = 0u, hi = 0xFFFFFFFFu;
  while (lo < hi) {
    unsigned int mid = (unsigned int)((((unsigned long long)lo + (unsigned long long)hi) + 1ull) >> 1);
    if (threadIdx.x == 0) cnt = 0;
    __syncthreads();
    int c = 0;
#pragma unroll
    for (int i = 0; i < 2; ++i)
      if (abits[threadIdx.x + 256 * i] >= mid) ++c;
    atomicAdd(&cnt, c);
    __syncthreads();
    if (cnt >= KK) lo = mid; else hi = mid - 1u;
    __syncthreads();
  }
  float thr = __uint_as_float(lo);   // exactly the K-th largest element
  float nv[2];
  float asum = 0.0f;
#pragma unroll
  for (int i = 0; i < 2; ++i) {
    float x = hv[i] + ((av[i] >= thr) ? rv[i] : 0.0f);
    nv[i] = x;
    asum += fabsf(x);
  }
  red[threadIdx.x] = asum;
  __syncthreads();
  for (int st = 128; st > 0; st >>= 1) {
    if ((int)threadIdx.x < st) red[threadIdx.x] += red[threadIdx.x + st];
    __syncthreads();
  }
  float inv = 1.0f / fmaxf(red[0], EPSF);
#pragma unroll
  for (int i = 0; i < 2; ++i)
    ht[((size_t)s * RR + r) * LL + threadIdx.x + 256 * i] = nv[i] * inv;
}

// ---------------- rs = ht_att @ seq via V_WMMA_F32_16X16X4_F32 ---------------
// One wave per 16x16 tile. A layout: lane m=L%16, K pair = (lane/16)*2.
// B/D rows striped across lanes within a VGPR (ISA 7.12.2).
__global__ __launch_bounds__(32) void k_rs_wmma(const float* __restrict__ ht,
                                                const float* __restrict__ seq,
                                                float* __restrict__ rs) {
  const int lane = threadIdx.x;
  const int tn = blockIdx.x, tm = blockIdx.y, s = blockIdx.z;
  const int klo = (lane >> 4) << 1;                       // 0 or 2
  const float* arow = ht + ((size_t)s * RR + tm * 16 + (lane & 15)) * LL;
  const float* bcol = seq + (size_t)s * LL * DD + tn * 16 + (lane & 15);
  v8f acc = {};
  for (int k = 0; k < LL; k += 4) {
    v2f a = *(const v2f*)(arow + k + klo);               // 8B aligned
    v2f b;
    b.x = bcol[(size_t)(k + klo) * DD];
    b.y = bcol[(size_t)(k + klo + 1) * DD];
    acc = __builtin_amdgcn_wmma_f32_16x16x4_f32(false, a, false, b,
                                                (short)0, acc, false, false);
  }
  const int rbase = tm * 16 + ((lane >> 4) << 3);
  float* orow = rs + (size_t)s * RR * DD + tn * 16 + (lane & 15);
#pragma unroll
  for (int vi = 0; vi < 8; ++vi)
    orow[(size_t)(rbase + vi) * DD] = acc[vi];
}

// ---------------- hs2/ts2: tanh( [ent|rs] @ W^T + b ) via WMMA ---------------
// blockIdx.z = s*2 + which (0: head/Wh, 1: tail/Wt). A gathers concat on the fly.
__global__ __launch_bounds__(32) void k_out_wmma(const float* __restrict__ ent,
                                                 const float* __restrict__ rs,
                                                 const int* __restrict__ hts,
                                                 const float* __restrict__ Wh,
                                                 const float* __restrict__ bh,
                                                 const float* __restrict__ Wt,
                                                 const float* __restrict__ bt,
                                                 float* __restrict__ out) {
  const int lane = threadIdx.x;
  const int tn = blockIdx.x, tm = blockIdx.y;
  const int s = blockIdx.z >> 1;
  const int which = blockIdx.z & 1;
  const float* W = which ? Wt : Wh;
  const float* bias = which ? bt : bh;
  const int mr = tm * 16 + (lane & 15);                   // pair row for this lane's A
  const int e = hts[((size_t)s * RR + mr) * 2 + which];
  const float* aent = ent + ((size_t)s * EE + e) * DD;    // k in [0,768)
  const float* ars = rs + ((size_t)s * RR + mr) * DD - DD;// k in [768,1536)
  const int klo = (lane >> 4) << 1;
  const int n = tn * 16 + (lane & 15);
  const float* wrow = W + (size_t)n * D2;                 // B[k][n] = W[n][k]
  v8f acc = {};
  for (int k = 0; k < D2; k += 4) {
    int kk = k + klo;                                     // pair never straddles 768
    v2f a;
    if (kk < DD) a = *(const v2f*)(aent + kk);
    else         a = *(const v2f*)(ars + kk);
    v2f b = *(const v2f*)(wrow + kk);
    acc = __builtin_amdgcn_wmma_f32_16x16x4_f32(false, a, false, b,
                                                (short)0, acc, false, false);
  }
  float bn = bias[n];
  const int rbase = tm * 16 + ((lane >> 4) << 3);
#pragma unroll
  for (int vi = 0; vi < 8; ++vi) {
    int r = rbase + vi;
    out[((size_t)s * RR + r) * D2 + which * DD + n] = tanhf(acc[vi] + bn);
  }
}

extern "C" void kernel_launch(void* const* d_in, const int* in_sizes, int n_in,
                              void* d_out, int out_size, void* d_ws, size_t ws_size,
                              hipStream_t stream) {
  (void)in_sizes; (void)n_in; (void)out_size; (void)ws_size;
  const float* seq     = (const float*)d_in[0];
  const float* att     = (const float*)d_in[1];
  const int*   msk     = (const int*)d_in[2];
  const int*   starts  = (const int*)d_in[3];
  const int*   hts     = (const int*)d_in[4];
  const float* rel_cls = (const float*)d_in[5];
  const float* Wq      = (const float*)d_in[6];
  const float* bq      = (const float*)d_in[7];
  const float* Wk      = (const float*)d_in[8];
  const float* bk      = (const float*)d_in[9];
  const float* Wh      = (const float*)d_in[10];
  const float* bh      = (const float*)d_in[11];
  const float* Wt      = (const float*)d_in[12];
  const float* bt      = (const float*)d_in[13];
  float* out = (float*)d_out;

  float* ws = (float*)d_ws;
  size_t off = 0;
  float* ent   = ws + off; off += (size_t)SS * EE * DD;        // 196608
  float* entA  = ws + off; off += (size_t)SS * EE * HH * LL;   // 1572864
  float* ht    = ws + off; off += (size_t)SS * RR * LL;        // 262144
  float* rel   = ws + off; off += (size_t)SS * LL;             // 4096
  float* qv    = ws + off; off += DD;
  float* vv    = ws + off; off += DD;
  float* cv    = ws + off; off += 64;
  float* rsbuf = ws + off; off += (size_t)SS * RR * DD;        // 393216

  k_q<<<3, 256, 0, stream>>>(Wq, bq, rel_cls, qv);
  k_v<<<3, 256, 0, stream>>>(Wk, qv, vv);
  k_c<<<1, 256, 0, stream>>>(bk, qv, cv);
  k_entemb<<<(SS * EE * DD + 255) / 256, 256, 0, stream>>>(seq, starts, ent);
  k_entatt<<<(SS * EE * HH * LL + 255) / 256, 256, 0, stream>>>(att, starts, entA);
  k_relatt<<<SS, 256, 0, stream>>>(seq, msk, vv, cv, rel);
  k_htatt<<<SS * RR, 256, 0, stream>>>(entA, hts, ht);
  k_topk<<<SS * RR, 256, 0, stream>>>(rel, ht);
  k_rs_wmma<<<dim3(DD / 16, RR / 16, SS), 32, 0, stream>>>(ht, seq, rsbuf);
  k_out_wmma<<<dim3(DD / 16, RR / 16, SS * 2), 32, 0, stream>>>(ent, rsbuf, hts,
                                                                Wh, bh, Wt, bt, out);
}